// construct_graph_61452392071868
// MI455X (gfx1250) — compile-verified
//
#include <hip/hip_runtime.h>
#include <hip/hip_bf16.h>

// ---------------------------------------------------------------------------
// Graph construction for point clouds (FPS + kNN + grouping), MI455X / gfx1250.
//
// Shapes (from reference): B=4, N=8192 points, C=64 feature channels,
// m=2048 FPS samples, k=20 neighbors.
// Outputs: FPS_xyz [B,m,3] then graph_features [B,3+C,m,k], concatenated flat.
//
// Key MI455X choices:
//  * FPS: one block per batch, all coords in LDS (96KB of the 320KB/WGP),
//    wave32 shuffle argmax + cross-wave LDS reduce. Zero HBM traffic in the
//    2047-iteration serial loop.
//  * kNN distances via V_WMMA_F32_16X16X4_F32: the K-dim is 3(+1 pad), which
//    is exactly the 16x16x4 f32 WMMA shape. Ranking-equivalent distance
//    D[m][n] = -2 q.p + |p|^2 from A=[qx,qy,qz,1], B=[-2px,-2py,-2pz,|p|^2].
//  * Top-20 per query kept in registers (fully unrolled bubble insert).
// ---------------------------------------------------------------------------

#define NB    4
#define NPTS  8192
#define NFPS  2048
#define NC    64
#define NK    20

typedef float v2f __attribute__((ext_vector_type(2)));
typedef float v8f __attribute__((ext_vector_type(8)));

// --------------------------------------------------------------------------
// Kernel 0: precompute B-matrix point vectors pts4[b][n] = {-2x,-2y,-2z,|p|^2}
// --------------------------------------------------------------------------
__global__ __launch_bounds__(256) void prep_pts4_kernel(
    const float* __restrict__ xyz, float* __restrict__ pts4) {
  int t = blockIdx.x * 256 + threadIdx.x;
  if (t >= NB * NPTS) return;
  float x = xyz[3 * t + 0];
  float y = xyz[3 * t + 1];
  float z = xyz[3 * t + 2];
  float4 v = make_float4(-2.0f * x, -2.0f * y, -2.0f * z, x * x + y * y + z * z);
  ((float4*)pts4)[t] = v;
}

// --------------------------------------------------------------------------
// Kernel 1: furthest point sampling. grid = B, block = 1024 (32 waves).
// Emits: center_idx[b][m], q4[b][m]={x,y,z,1}, FPS_xyz -> d_out head.
// Matches reference: idx[0]=0; idx[t]=argmax(min_d) after folding idx[t-1].
// Ties -> lowest index (jnp.argmax takes first max).
// --------------------------------------------------------------------------
__global__ __launch_bounds__(1024) void fps_kernel(
    const float* __restrict__ xyz, float* __restrict__ fps_xyz_out,
    int* __restrict__ center_idx, float* __restrict__ q4) {
  __shared__ float sx[NPTS];
  __shared__ float sy[NPTS];
  __shared__ float sz[NPTS];
  __shared__ float swv[32];
  __shared__ int   swi[32];
  __shared__ int   ssel;

  const int b   = blockIdx.x;
  const int tid = threadIdx.x;
  const int lane = tid & 31;
  const int wave = tid >> 5;
  const float* xb = xyz + (size_t)b * NPTS * 3;

  for (int i = tid; i < NPTS; i += 1024) {
    sx[i] = xb[3 * i + 0];
    sy[i] = xb[3 * i + 1];
    sz[i] = xb[3 * i + 2];
  }
  if (tid == 0) ssel = 0;
  __syncthreads();

  // Each thread owns 8 points: p = tid + i*1024 (conflict-free LDS banks).
  float px[8], py[8], pz[8], mind[8];
#pragma unroll
  for (int i = 0; i < 8; ++i) {
    int p = tid + i * 1024;
    px[i] = sx[p]; py[i] = sy[p]; pz[i] = sz[p];
    mind[i] = 1e10f;
  }

  if (tid == 0) {  // t = 0: seed index 0
    center_idx[b * NFPS + 0] = 0;
    fps_xyz_out[(size_t)b * NFPS * 3 + 0] = sx[0];
    fps_xyz_out[(size_t)b * NFPS * 3 + 1] = sy[0];
    fps_xyz_out[(size_t)b * NFPS * 3 + 2] = sz[0];
    ((float4*)q4)[b * NFPS + 0] = make_float4(sx[0], sy[0], sz[0], 1.0f);
  }

  for (int t = 1; t < NFPS; ++t) {
    const int sel = ssel;                       // broadcast LDS read
    const float lx = sx[sel], ly = sy[sel], lz = sz[sel];

    float bv = -1.0f;
    int   bi = 0;
#pragma unroll
    for (int i = 0; i < 8; ++i) {
      float dx = px[i] - lx, dy = py[i] - ly, dz = pz[i] - lz;
      float d = dx * dx + dy * dy + dz * dz;
      float mn = fminf(mind[i], d);
      mind[i] = mn;
      if (mn > bv) { bv = mn; bi = tid + i * 1024; }  // strict > keeps lowest p
    }
    // wave32 (val,idx) max-reduce, lowest index on tie
#pragma unroll
    for (int off = 16; off > 0; off >>= 1) {
      float ov = __shfl_xor(bv, off, 32);
      int   oi = __shfl_xor(bi, off, 32);
      if (ov > bv || (ov == bv && oi < bi)) { bv = ov; bi = oi; }
    }
    if (lane == 0) { swv[wave] = bv; swi[wave] = bi; }
    __syncthreads();
    if (wave == 0) {
      float v = swv[lane];
      int   ii = swi[lane];
#pragma unroll
      for (int off = 16; off > 0; off >>= 1) {
        float ov = __shfl_xor(v, off, 32);
        int   oi = __shfl_xor(ii, off, 32);
        if (ov > v || (ov == v && oi < ii)) { v = ov; ii = oi; }
      }
      if (lane == 0) {
        ssel = ii;
        center_idx[b * NFPS + t] = ii;
        float wx = sx[ii], wy = sy[ii], wz = sz[ii];
        size_t o = ((size_t)b * NFPS + t) * 3;
        fps_xyz_out[o + 0] = wx;
        fps_xyz_out[o + 1] = wy;
        fps_xyz_out[o + 2] = wz;
        ((float4*)q4)[b * NFPS + t] = make_float4(wx, wy, wz, 1.0f);
      }
    }
    __syncthreads();
  }
}

// --------------------------------------------------------------------------
// Kernel 2: WMMA-tiled kNN. grid = B*16 blocks, block = 128 threads (4 waves).
// Block handles 128 queries; points swept in 64-wide chunks.
// Wave w computes row-tiles {2w, 2w+1} x 4 col-tiles with
// V_WMMA_F32_16X16X4_F32, stores 16x16 tiles into LDS (row stride 69 floats,
// bank-disjoint for the two half-wave row groups), then thread t scans row t
// maintaining a register-resident sorted top-20 (ascending distance).
// --------------------------------------------------------------------------
__global__ __launch_bounds__(128) void knn_kernel(
    const float* __restrict__ q4, const float* __restrict__ pts4,
    int* __restrict__ knn_idx) {
  __shared__ float sd[128 * 69];

  const int tid  = threadIdx.x;
  const int lane = tid & 31;
  const int wave = tid >> 5;
  const int b    = blockIdx.x >> 4;          // 16 query blocks per batch
  const int qblk = (blockIdx.x & 15) * 128;

  const int mrow = lane & 15;                // M (A) or N (B) within tile
  const int kk   = (lane >> 4) * 2;          // K pair selected by half-wave

  // A fragments (per ISA: VGPR0 = K{0|2}, VGPR1 = K{1|3}; M = lane%16).
  const int rt0 = wave * 2, rt1 = wave * 2 + 1;
  const float* qb = q4 + ((size_t)b * NFPS + qblk) * 4;
  v2f a0, a1;
  a0.x = qb[(rt0 * 16 + mrow) * 4 + kk];
  a0.y = qb[(rt0 * 16 + mrow) * 4 + kk + 1];
  a1.x = qb[(rt1 * 16 + mrow) * 4 + kk];
  a1.y = qb[(rt1 * 16 + mrow) * 4 + kk + 1];

  const float* pb = pts4 + (size_t)b * NPTS * 4;

  float tv[NK];
  int   ti[NK];
#pragma unroll
  for (int j = 0; j < NK; ++j) { tv[j] = 3.0e38f; ti[j] = 0; }

  for (int pc = 0; pc < NPTS; pc += 64) {
#pragma unroll
    for (int ct = 0; ct < 4; ++ct) {
      const int col0 = pc + ct * 16;
      v2f bf;  // B fragment: same lane pattern as A (K striped by half-wave)
      bf.x = pb[(col0 + mrow) * 4 + kk];
      bf.y = pb[(col0 + mrow) * 4 + kk + 1];
      v8f z = {};
      v8f d0 = __builtin_amdgcn_wmma_f32_16x16x4_f32(
          false, a0, false, bf, (short)0, z, false, false);
      v8f d1 = __builtin_amdgcn_wmma_f32_16x16x4_f32(
          false, a1, false, bf, (short)0, z, false, false);
      // D layout: VGPR r -> row (r + 8*(lane>=16)), col = lane%16.
      const int cb = ct * 16 + mrow;
      const int r0 = rt0 * 16 + (lane >> 4) * 8;
      const int r1 = rt1 * 16 + (lane >> 4) * 8;
#pragma unroll
      for (int r = 0; r < 8; ++r) sd[(r0 + r) * 69 + cb] = d0[r];
#pragma unroll
      for (int r = 0; r < 8; ++r) sd[(r1 + r) * 69 + cb] = d1[r];
    }
    __syncthreads();

    // Thread t scans its query row (stride-69 rows -> conflict-free columns).
    const float* row = &sd[tid * 69];
    for (int j = 0; j < 64; ++j) {
      float d = row[j];
      if (d < tv[NK - 1]) {
        tv[NK - 1] = d;
        ti[NK - 1] = pc + j;
#pragma unroll
        for (int s = NK - 1; s > 0; --s) {
          if (tv[s] < tv[s - 1]) {   // strict: stable on ties (lower idx first)
            float tf = tv[s]; tv[s] = tv[s - 1]; tv[s - 1] = tf;
            int   tn = ti[s]; ti[s] = ti[s - 1]; ti[s - 1] = tn;
          }
        }
      }
    }
    __syncthreads();
  }

  int* out = knn_idx + ((size_t)b * NFPS + qblk + tid) * NK;
#pragma unroll
  for (int j = 0; j < NK; ++j) out[j] = ti[j];
}

// --------------------------------------------------------------------------
// Kernel 3: gather + center-subtract + write graph_features [B,3+C,m,k].
// One thread per (b,m,k); k is the fastest thread dim -> coalesced writes.
// Scattered feature reads stay L2-resident (features = 8MB << 192MB L2).
// --------------------------------------------------------------------------
__global__ __launch_bounds__(256) void gather_kernel(
    const float* __restrict__ xyz, const float* __restrict__ feats,
    const int* __restrict__ center_idx, const int* __restrict__ knn_idx,
    const float* __restrict__ q4, float* __restrict__ graph_out) {
  int t = blockIdx.x * 256 + threadIdx.x;
  if (t >= NB * NFPS * NK) return;
  const int k = t % NK;
  const int m = (t / NK) % NFPS;
  const int b = t / (NK * NFPS);

  const int p  = knn_idx[((size_t)b * NFPS + m) * NK + k];
  const int ci = center_idx[b * NFPS + m];

  const float4 q = ((const float4*)q4)[b * NFPS + m];  // center coords
  const float* xp = xyz + ((size_t)b * NPTS + p) * 3;

  const size_t cs = (size_t)NFPS * NK;  // channel stride in output
  float* ob = graph_out + (size_t)b * (3 + NC) * cs + (size_t)m * NK + k;

  ob[0 * cs] = xp[0] - q.x;
  ob[1 * cs] = xp[1] - q.y;
  ob[2 * cs] = xp[2] - q.z;

  const float* fb = feats + (size_t)b * NC * NPTS;
#pragma unroll 4
  for (int c = 0; c < NC; ++c) {
    ob[(size_t)(3 + c) * cs] = fb[(size_t)c * NPTS + p] - fb[(size_t)c * NPTS + ci];
  }
}

// --------------------------------------------------------------------------
// Launch
// --------------------------------------------------------------------------
extern "C" void kernel_launch(void* const* d_in, const int* in_sizes, int n_in,
                              void* d_out, int out_size, void* d_ws, size_t ws_size,
                              hipStream_t stream) {
  const float* xyz   = (const float*)d_in[0];   // [4,8192,3]
  const float* feats = (const float*)d_in[1];   // [4,64,8192]
  float* out = (float*)d_out;
  float* fps_xyz_out = out;                     // [4,2048,3]
  float* graph_out   = out + (size_t)NB * NFPS * 3;  // [4,67,2048,20]

  // Workspace layout (16B-aligned offsets):
  char* ws = (char*)d_ws;
  int*   center_idx = (int*)(ws + 0);             //  32768 B  (B*m ints)
  int*   knn_idx    = (int*)(ws + 32768);         // 655360 B  (B*m*k ints)
  float* q4         = (float*)(ws + 688128);      // 131072 B  (B*m*4 f32)
  float* pts4       = (float*)(ws + 819200);      // 524288 B  (B*N*4 f32)

  prep_pts4_kernel<<<(NB * NPTS + 255) / 256, 256, 0, stream>>>(xyz, pts4);
  fps_kernel<<<NB, 1024, 0, stream>>>(xyz, fps_xyz_out, center_idx, q4);
  knn_kernel<<<NB * 16, 128, 0, stream>>>(q4, pts4, knn_idx);
  gather_kernel<<<(NB * NFPS * NK + 255) / 256, 256, 0, stream>>>(
      xyz, feats, center_idx, knn_idx, q4, graph_out);
}